// StateTrajectoryNetworkFF_58755152609972
// MI455X (gfx1250) — compile-verified
//
#include <hip/hip_runtime.h>
#include <stdint.h>

// StateTrajectoryNetwork free-running recurrence, MI455X (gfx1250).
// Latency-bound serial scan: 1024 independent chains x 16384 steps.
// One batch element per lane, wave32, one wave per workgroup.
// x staged through LDS with async-to-LDS B128 loads (double buffered),
// tanh via gfx1250 hardware v_tanh_f32.

#define SEQ    16384
#define BATCH  1024
#define TPB    32
#define CHUNK  64
#define NCHUNK (SEQ / CHUNK)   // 256

typedef __attribute__((ext_vector_type(4))) int v4i;
typedef __attribute__((address_space(1))) v4i glob_v4i;   // global (AS1)
typedef __attribute__((address_space(3))) v4i lds_v4i;    // LDS (AS3)

__device__ __forceinline__ void async_cp16(const float* g, float* l) {
#if __has_builtin(__builtin_amdgcn_global_load_async_to_lds_b128)
  // ROCm 7.2 signature: (v4i AS1* gsrc, v4i AS3* ldst, imm offset, imm cpol).
  // Direct C-style addrspace casts: flat->AS1 identity, flat->AS3 keeps the
  // low-32-bit LDS offset.
  __builtin_amdgcn_global_load_async_to_lds_b128(
      (glob_v4i*)g,
      (lds_v4i*)l,
      /*imm_offset=*/0, /*cpol=*/0);
#else
  *(float4*)l = *(const float4*)g;   // fallback: sync copy via VGPRs
#endif
}

__device__ __forceinline__ void async_wait0() {
#if __has_builtin(__builtin_amdgcn_s_wait_asynccnt)
  __builtin_amdgcn_s_wait_asynccnt(0);
#else
  asm volatile("s_wait_asynccnt 0x0" ::: "memory");
#endif
  asm volatile("" ::: "memory");     // keep ds_loads ordered after the wait
}

__device__ __forceinline__ float htanh(float v) {
#if __has_builtin(__builtin_amdgcn_tanhf)
  return __builtin_amdgcn_tanhf(v);  // v_tanh_f32 (gfx1250 TRANS32)
#else
  float r;
  asm("v_tanh_f32 %0, %1\n\tv_nop" : "=v"(r) : "v"(v));  // v_nop: TRANS hazard
  return r;
#endif
}

__global__ void __launch_bounds__(TPB, 1)
stn_recurrence(const float* __restrict__ x,
               const float* __restrict__ W1,
               const float* __restrict__ W2,
               const float* __restrict__ b2v,
               const float* __restrict__ W3,
               const float* __restrict__ W4,
               float* __restrict__ out)
{
  __shared__ float xs[2][CHUNK][TPB];   // 16 KB of the 320 KB WGP LDS

  const int lane = threadIdx.x;           // batch element within block
  const int b0   = blockIdx.x * TPB;
  const int b    = b0 + lane;

  // Uniform weights -> SGPRs via scalar loads.
  float w1[8], w2[16], bb[4], w3[16], w4[4];
#pragma unroll
  for (int i = 0; i < 8;  ++i) w1[i] = W1[i];
#pragma unroll
  for (int i = 0; i < 16; ++i) w2[i] = W2[i];
#pragma unroll
  for (int i = 0; i < 4;  ++i) bb[i] = b2v[i];
#pragma unroll
  for (int i = 0; i < 16; ++i) w3[i] = W3[i];
#pragma unroll
  for (int i = 0; i < 4;  ++i) w4[i] = W4[i];

  // Async staging pattern: each B128 issue moves 4 time-rows of this block's
  // 32-float column slice (32 lanes x 16B = 512B = 4 rows x 128B).
  const int row_off = lane >> 3;          // 0..3
  const int col4    = (lane & 7) << 2;    // 0,4,...,28
  const float* gsrc = x + (size_t)row_off * BATCH + b0 + col4;
  float* lrow0 = &xs[0][row_off][col4];
  float* lrow1 = &xs[1][row_off][col4];

  // Prologue: stage chunk 0 into buffer 0.
#pragma unroll
  for (int i = 0; i < CHUNK / 4; ++i)
    async_cp16(gsrc + (size_t)i * 4 * BATCH, lrow0 + i * 4 * TPB);

  float s = 0.0f;                         // recurrent scalar state
  for (int c = 0; c < NCHUNK; ++c) {
    async_wait0();                        // chunk c resident in LDS
    if (c + 1 < NCHUNK) {                 // overlap: stage chunk c+1
      const float* g = gsrc + (size_t)(c + 1) * CHUNK * BATCH;
      float* l = ((c + 1) & 1) ? lrow1 : lrow0;
#pragma unroll
      for (int i = 0; i < CHUNK / 4; ++i)
        async_cp16(g + (size_t)i * 4 * BATCH, l + i * 4 * TPB);
    }

    const float (*buf)[TPB] = xs[c & 1];
    float* op = out + (size_t)c * CHUNK * BATCH + b;

#pragma unroll 8
    for (int t = 0; t < CHUNK; ++t) {
      float xv = buf[t][lane];            // ds_load_b32, conflict-free

      // layer 1: tanh(W1 @ [x, s])        chain: mul+fma+tanh
      float a0 = htanh(fmaf(w1[0], xv, w1[1] * s));
      float a1 = htanh(fmaf(w1[2], xv, w1[3] * s));
      float a2 = htanh(fmaf(w1[4], xv, w1[5] * s));
      float a3 = htanh(fmaf(w1[6], xv, w1[7] * s));

      // layer 2: tanh(W2 @ a + b)         FMA tree, depth 3
      float u0 = htanh(fmaf(w2[ 0], a0, fmaf(w2[ 1], a1, bb[0])) + fmaf(w2[ 2], a2, w2[ 3] * a3));
      float u1 = htanh(fmaf(w2[ 4], a0, fmaf(w2[ 5], a1, bb[1])) + fmaf(w2[ 6], a2, w2[ 7] * a3));
      float u2 = htanh(fmaf(w2[ 8], a0, fmaf(w2[ 9], a1, bb[2])) + fmaf(w2[10], a2, w2[11] * a3));
      float u3 = htanh(fmaf(w2[12], a0, fmaf(w2[13], a1, bb[3])) + fmaf(w2[14], a2, w2[15] * a3));

      // layer 3: tanh(W3 @ u)
      float g0 = htanh(fmaf(w3[ 0], u0, w3[ 1] * u1) + fmaf(w3[ 2], u2, w3[ 3] * u3));
      float g1 = htanh(fmaf(w3[ 4], u0, w3[ 5] * u1) + fmaf(w3[ 6], u2, w3[ 7] * u3));
      float g2 = htanh(fmaf(w3[ 8], u0, w3[ 9] * u1) + fmaf(w3[10], u2, w3[11] * u3));
      float g3 = htanh(fmaf(w3[12], u0, w3[13] * u1) + fmaf(w3[14], u2, w3[15] * u3));

      // layer 4 + residual state update
      s += fmaf(w4[0], g0, w4[1] * g1) + fmaf(w4[2], g2, w4[3] * g3);

      op[(size_t)t * BATCH] = s;          // coalesced, fire-and-forget store
    }
  }
}

extern "C" void kernel_launch(void* const* d_in, const int* in_sizes, int n_in,
                              void* d_out, int out_size, void* d_ws, size_t ws_size,
                              hipStream_t stream) {
  (void)in_sizes; (void)n_in; (void)out_size; (void)d_ws; (void)ws_size;
  const float* x  = (const float*)d_in[0];
  const float* W1 = (const float*)d_in[1];
  const float* W2 = (const float*)d_in[2];
  const float* b2 = (const float*)d_in[3];
  const float* W3 = (const float*)d_in[4];
  const float* W4 = (const float*)d_in[5];
  float* out = (float*)d_out;

  stn_recurrence<<<dim3(BATCH / TPB), dim3(TPB), 0, stream>>>(
      x, W1, W2, b2, W3, W4, out);
}